// ChamferLoss_24077586661573
// MI455X (gfx1250) — compile-verified
//
#include <hip/hip_runtime.h>
#include <stdint.h>

typedef __attribute__((ext_vector_type(2))) float v2f;
typedef __attribute__((ext_vector_type(8))) float v8f;

#define B_ 2
#define N_ 8192
#define M_ 8192
#define ALPHA_ 0.5f

#define WAVES_PER_WG 4
#define ROWS_PER_WAVE 32                      // two 16-row WMMA tiles per wave
#define ROWS_PER_WG (ROWS_PER_WAVE * WAVES_PER_WG)  // 128
#define WG_PER_BATCH (N_ / ROWS_PER_WG)       // 64
#define COL_SPLIT 2
#define COLS_PER_WG (M_ / COL_SPLIT)          // 4096
#define CT (COLS_PER_WG / 16)                 // 256 column tiles per WG

static __device__ __forceinline__ unsigned long long pack_fi(float v, unsigned idx) {
    return (((unsigned long long)__float_as_uint(v)) << 32) | (unsigned long long)idx;
}

static __device__ __forceinline__ float clamp0(float v) {
    // v_med3_f32(v, 0, +inf) == max(v, 0) in one instruction (no canonicalize)
    return __builtin_amdgcn_fmed3f(v, 0.0f, __builtin_inff());
}

static __device__ __forceinline__ unsigned long long shfl_xor_u64(unsigned long long v, int mask) {
    unsigned lo = (unsigned)(v & 0xFFFFFFFFull);
    unsigned hi = (unsigned)(v >> 32);
    lo = (unsigned)__shfl_xor((int)lo, mask, 32);
    hi = (unsigned)__shfl_xor((int)hi, mask, 32);
    return (((unsigned long long)hi) << 32) | (unsigned long long)lo;
}

static __device__ __forceinline__ unsigned long long umin64(unsigned long long a, unsigned long long b) {
    return a < b ? a : b;
}

// ---------------------------------------------------------------------------
// Kernel 0: init workspace (min tables = +inf packed, norm bits = 0, sums = 0)
// ---------------------------------------------------------------------------
__global__ void chamfer_init_kernel(unsigned long long* __restrict__ colMinG,
                                    unsigned long long* __restrict__ rowMinG,
                                    unsigned* __restrict__ normBits,
                                    float* __restrict__ sums) {
    int i = blockIdx.x * blockDim.x + threadIdx.x;
    if (i < B_ * M_) colMinG[i] = ~0ull;
    if (i < B_ * N_) rowMinG[i] = ~0ull;
    if (i < 4) sums[i] = 0.0f;
    if (i < B_) normBits[i] = 0u;
}

// ---------------------------------------------------------------------------
// Kernel 1: build WMMA-ready fragments + per-batch max ||x||.
//   prepX[i] = [x0, x1, x2, x^2]       (A row, K=0..3; K=3 carries x^2)
//   prepY[i] = [-2y0, -2y1, -2y2, y^2] (B col, K=0..2 scaled; .w feeds C)
// ---------------------------------------------------------------------------
__global__ void chamfer_prep_kernel(const float* __restrict__ x,
                                    const float* __restrict__ y,
                                    float4* __restrict__ prepX,
                                    float4* __restrict__ prepY,
                                    unsigned* __restrict__ normBits) {
    int i = blockIdx.x * blockDim.x + threadIdx.x;
    if (i < B_ * N_) {
        const float* xp = x + (size_t)i * 6;
        float x0 = xp[0], x1 = xp[1], x2 = xp[2];
        float xs = x0 * x0 + x1 * x1 + x2 * x2;
        prepX[i] = make_float4(x0, x1, x2, xs);
        atomicMax(&normBits[i / N_], __float_as_uint(sqrtf(xs)));
    }
    if (i < B_ * M_) {
        const float* yp = y + (size_t)i * 6;
        float y0 = yp[0], y1 = yp[1], y2 = yp[2];
        float ys = y0 * y0 + y1 * y1 + y2 * y2;
        prepY[i] = make_float4(-2.0f * y0, -2.0f * y1, -2.0f * y2, ys);
    }
}

// ---------------------------------------------------------------------------
// Kernel 2: main WMMA distance + min/argmin kernel.
// Each wave owns 32 rows (2 WMMA tiles); a workgroup (4 waves, 128 rows) sweeps
// 4096 columns (COL_SPLIT=2). Per 16-col tile:
//   d2 = V_WMMA_F32_16X16X4_F32(A=[x,x^2], B=[-2y,1], C=y^2 broadcast)
// Row mins: packed u64 in registers, lane-reduced once at the end, merged with
// global_atomic_min_u64. Column mins: tree-reduced per lane, then every lane
// atomic-mins its column's LDS slot (lanes L and L+16 share a column, the
// ds_min_u64 does the final combine).
// ---------------------------------------------------------------------------
__global__ __launch_bounds__(128) void chamfer_wmma_min_kernel(
    const float4* __restrict__ prepX, const float4* __restrict__ prepY,
    unsigned long long* __restrict__ rowMinG,
    unsigned long long* __restrict__ colMinG) {
    __shared__ unsigned long long ldsCol[COLS_PER_WG];  // 32 KB

    const int tid = threadIdx.x;
    const int lane = tid & 31;
    const int wave = tid >> 5;

    const int bid = blockIdx.x;
    const int b = bid / (WG_PER_BATCH * COL_SPLIT);
    const int rem = bid % (WG_PER_BATCH * COL_SPLIT);
    const int wg = rem / COL_SPLIT;
    const int split = rem % COL_SPLIT;

    const int rowbase = wg * ROWS_PER_WG + wave * ROWS_PER_WAVE;
    const int colbase = split * COLS_PER_WG;

    for (int i = tid; i < COLS_PER_WG; i += blockDim.x) ldsCol[i] = ~0ull;
    __syncthreads();

    const float4* pXb = prepX + (size_t)b * N_;
    const float4* pYb = prepY + (size_t)b * M_;

    const int r = lane & 15;
    const bool hi = lane >= 16;

    // A fragments (16x4 f32: lanes 0-15 hold K=0,1 of row=lane; 16-31 hold K=2,3)
    const float4 ax0 = pXb[rowbase + r];
    const float4 ax1 = pXb[rowbase + 16 + r];
    v2f a0, a1;
    a0.x = hi ? ax0.z : ax0.x;
    a0.y = hi ? ax0.w : ax0.y;
    a1.x = hi ? ax1.z : ax1.x;
    a1.y = hi ? ax1.w : ax1.y;

    unsigned long long runRow[16];
#pragma unroll
    for (int v = 0; v < 16; ++v) runRow[v] = ~0ull;

    const unsigned rowG0 = (unsigned)(rowbase + (hi ? 8 : 0));
    const unsigned rowG1 = rowG0 + 16u;

    // distance-2 software pipeline on the B-fragment loads
    float4 p0 = pYb[colbase + r];
    float4 p1 = pYb[colbase + 16 + r];
#pragma unroll 2
    for (int mt = 0; mt < CT; ++mt) {
        const int colLocal = mt * 16 + r;
        const unsigned colGlobal = (unsigned)(colbase + colLocal);
        const int pfLocal = (mt + 2 < CT) ? (mt + 2) * 16 + r : r;
        const float4 pn = pYb[colbase + pfLocal];  // prefetch tile mt+2

        // B fragment (4x16 f32: lanes 0-15 hold K=0,1 of col=lane; 16-31 K=2,3)
        v2f bf;
        bf.x = hi ? p0.z : p0.x;
        bf.y = hi ? 1.0f : p0.y;
        v8f c;
#pragma unroll
        for (int v = 0; v < 8; ++v) c[v] = p0.w;  // per-column y^2 broadcast

        v8f d0 = __builtin_amdgcn_wmma_f32_16x16x4_f32(
            false, a0, false, bf, (short)0, c, false, false);
        v8f d1 = __builtin_amdgcn_wmma_f32_16x16x4_f32(
            false, a1, false, bf, (short)0, c, false, false);

        unsigned long long pk[16];
#pragma unroll
        for (int v = 0; v < 8; ++v) {
            float dv0 = clamp0(d0[v]);  // single v_med3_f32
            float dv1 = clamp0(d1[v]);
            runRow[v] = umin64(runRow[v], pack_fi(dv0, colGlobal));
            runRow[8 + v] = umin64(runRow[8 + v], pack_fi(dv1, colGlobal));
            pk[v] = pack_fi(dv0, rowG0 + (unsigned)v);
            pk[8 + v] = pack_fi(dv1, rowG1 + (unsigned)v);
        }
        // tree-reduce the 16 column candidates (depth 4, dual-issue friendly)
#pragma unroll
        for (int s = 8; s >= 1; s >>= 1) {
#pragma unroll
            for (int v = 0; v < s; ++v) pk[v] = umin64(pk[v], pk[v + s]);
        }
        // all 32 lanes: lanes L and L+16 share a column; ds_min_u64 combines.
        atomicMin(&ldsCol[colLocal], pk[0]);

        p0 = p1;
        p1 = pn;
    }

    // finish row mins: reduce across the 16 columns held by each 16-lane half
#pragma unroll
    for (int v = 0; v < 16; ++v) {
        unsigned long long t = runRow[v];
        t = umin64(t, shfl_xor_u64(t, 1));
        t = umin64(t, shfl_xor_u64(t, 2));
        t = umin64(t, shfl_xor_u64(t, 4));
        t = umin64(t, shfl_xor_u64(t, 8));
        runRow[v] = t;
    }
    if (lane == 0 || lane == 16) {
        const int rbase = rowbase + (hi ? 8 : 0);
#pragma unroll
        for (int v = 0; v < 8; ++v) {
            atomicMin(&rowMinG[(size_t)b * N_ + rbase + v], runRow[v]);
            atomicMin(&rowMinG[(size_t)b * N_ + rbase + 16 + v], runRow[8 + v]);
        }
    }

    // flush workgroup's column mins to global once
    __syncthreads();
    for (int i = tid; i < COLS_PER_WG; i += blockDim.x)
        atomicMin(&colMinG[(size_t)b * M_ + colbase + i], ldsCol[i]);
}

// ---------------------------------------------------------------------------
// Kernel 3: finalize — unpack mins, gather NN colors, block-reduce the 4 sums.
// ---------------------------------------------------------------------------
__global__ __launch_bounds__(256) void chamfer_finalize_kernel(
    const float* __restrict__ x, const float* __restrict__ y,
    const unsigned long long* __restrict__ rowMinG,
    const unsigned long long* __restrict__ colMinG,
    const unsigned* __restrict__ normBits, float* __restrict__ sums) {
    __shared__ float s[4];
    if (threadIdx.x < 4) s[threadIdx.x] = 0.0f;
    __syncthreads();

    int i = blockIdx.x * blockDim.x + threadIdx.x;
    if (i < B_ * N_) {
        int b = i / N_;
        int n = i % N_;
        unsigned long long p = rowMinG[i];
        float d2 = __uint_as_float((unsigned)(p >> 32));
        unsigned m = (unsigned)(p & 0xFFFFFFFFull);
        float nrm = __uint_as_float(normBits[b]);
        float dist = sqrtf(d2) / nrm;
        const float* cx = x + ((size_t)b * N_ + n) * 6 + 3;
        const float* cy = y + ((size_t)b * M_ + m) * 6 + 3;
        float c0 = cx[0] - cy[0], c1 = cx[1] - cy[1], c2 = cx[2] - cy[2];
        float cd = sqrtf(c0 * c0 + c1 * c1 + c2 * c2);
        atomicAdd(&s[0], dist);
        atomicAdd(&s[2], cd);
    } else if (i < B_ * N_ + B_ * M_) {
        int j = i - B_ * N_;
        int b = j / M_;
        int m = j % M_;
        unsigned long long p = colMinG[j];
        float d2 = __uint_as_float((unsigned)(p >> 32));
        unsigned n = (unsigned)(p & 0xFFFFFFFFull);
        float nrm = __uint_as_float(normBits[b]);
        float dist = sqrtf(d2) / nrm;
        const float* cy = y + ((size_t)b * M_ + m) * 6 + 3;
        const float* cx = x + ((size_t)b * N_ + n) * 6 + 3;
        float c0 = cy[0] - cx[0], c1 = cy[1] - cx[1], c2 = cy[2] - cx[2];
        float cd = sqrtf(c0 * c0 + c1 * c1 + c2 * c2);
        atomicAdd(&s[1], dist);
        atomicAdd(&s[3], cd);
    }

    __syncthreads();
    if (threadIdx.x < 4) atomicAdd(&sums[threadIdx.x], s[threadIdx.x]);
}

// ---------------------------------------------------------------------------
// Kernel 4: emit the 3 scalar outputs.
// ---------------------------------------------------------------------------
__global__ void chamfer_output_kernel(const float* __restrict__ sums,
                                      float* __restrict__ out) {
    if (blockIdx.x == 0 && threadIdx.x == 0) {
        float lx = sums[0] / (float)(B_ * N_);
        float ly = sums[1] / (float)(B_ * M_);
        float lc = sums[2] / (float)(B_ * N_) + sums[3] / (float)(B_ * M_);
        float geo = ALPHA_ * lx + ALPHA_ * ly;
        float col = (1.0f - ALPHA_) * lc;
        out[0] = geo + col;
        out[1] = geo;
        out[2] = col;
    }
}

extern "C" void kernel_launch(void* const* d_in, const int* in_sizes, int n_in,
                              void* d_out, int out_size, void* d_ws, size_t ws_size,
                              hipStream_t stream) {
    (void)in_sizes; (void)n_in; (void)out_size; (void)ws_size;
    const float* x = (const float*)d_in[0];  // (B, N, 6) f32
    const float* y = (const float*)d_in[1];  // (B, M, 6) f32
    float* out = (float*)d_out;              // 3 floats

    // Workspace carve-up (~786 KB)
    unsigned long long* colMinG = (unsigned long long*)d_ws;      // B*M u64
    unsigned long long* rowMinG = colMinG + (size_t)B_ * M_;      // B*N u64
    unsigned* normBits = (unsigned*)(rowMinG + (size_t)B_ * N_);  // 4 u32
    float* sums = (float*)(normBits + 4);                         // 4 f32
    float4* prepX = (float4*)(sums + 4);                          // B*N float4
    float4* prepY = prepX + (size_t)B_ * N_;                      // B*M float4

    const int threads = 256;

    chamfer_init_kernel<<<(B_ * M_ + threads - 1) / threads, threads, 0, stream>>>(
        colMinG, rowMinG, normBits, sums);

    chamfer_prep_kernel<<<(B_ * M_ + threads - 1) / threads, threads, 0, stream>>>(
        x, y, prepX, prepY, normBits);

    chamfer_wmma_min_kernel<<<B_ * WG_PER_BATCH * COL_SPLIT, 128, 0, stream>>>(
        prepX, prepY, rowMinG, colMinG);

    chamfer_finalize_kernel<<<(B_ * N_ + B_ * M_ + threads - 1) / threads, threads, 0,
                              stream>>>(x, y, rowMinG, colMinG, normBits, sums);

    chamfer_output_kernel<<<1, 1, 0, stream>>>(sums, out);
}